// SemiCRF_19009525252813
// MI455X (gfx1250) — compile-verified
//
#include <hip/hip_runtime.h>

#define TT 500
#define LL 26
#define KK 32
#define QQ 31          // K-1 usable history rows
#define IMPF 0.02f
#define NEGF (-1e30f)

typedef __attribute__((ext_vector_type(2))) float v2f;
typedef __attribute__((ext_vector_type(8))) float v8f;
typedef __attribute__((ext_vector_type(4))) unsigned int v4u;
typedef __attribute__((ext_vector_type(4))) int v4i;
typedef __attribute__((ext_vector_type(8))) int v8i;

// TDM: async-load a 31x26 f32 tile (row stride TT*LL elements) into LDS.
// 2D descriptor: groups 2/3 zero. Issued wave-uniform; tracked by TENSORcnt.
__device__ __forceinline__ void tdm_load_tile31x26(const float* gsrc,
                                                   unsigned lds_off) {
  unsigned long long ga = (unsigned long long)(uintptr_t)gsrc;
  v4u g0;
  g0[0] = 1u;                                         // count=1, user D#
  g0[1] = lds_off;                                    // lds_addr (bytes)
  g0[2] = (unsigned)(ga & 0xffffffffull);             // global_addr lo
  g0[3] = (unsigned)((ga >> 32) & 0x01ffffffull)      // global_addr hi (57b)
        | 0x80000000u;                                // type=2 ("image")
  v8i g1;
  g1[0] = 0x00020000;                                 // data_size=2 -> 4 bytes
  g1[1] = (int)(26u << 16);                           // tensor_dim0[15:0]
  g1[2] = (int)(31u << 16);                           // tensor_dim1[15:0]
  g1[3] = (int)(26u << 16);                           // tile_dim0
  g1[4] = 31;                                         // tile_dim1
  g1[5] = TT * LL;                                    // tensor_dim0_stride
  g1[6] = 0;
  g1[7] = 0;
  v4i z4 = {0, 0, 0, 0};
#if defined(__clang_major__) && (__clang_major__ >= 23)
  v8i z8 = {0, 0, 0, 0, 0, 0, 0, 0};
  __builtin_amdgcn_tensor_load_to_lds(g0, g1, z4, z4, z8, 0);
#else
  __builtin_amdgcn_tensor_load_to_lds(g0, g1, z4, z4, 0);
#endif
}

// One workgroup per batch element; 128 threads = 4 wave32.
// Ring of cached M-rows: M[q,l] = m_q + log(sum_lp exp(alpha_row[lp]-m_q)*E[lp,l])
// computed once per new alpha row via a 1x32 @ 32x32 FP32 WMMA matvec.
__launch_bounds__(128, 1)
__global__ void semicrf_fwd_kernel(const float* __restrict__ seg,
                                   const float* __restrict__ trans,
                                   float* __restrict__ out) {
  const int b    = blockIdx.x;
  const int tid  = threadIdx.x;
  const int lane = tid & 31;
  const int wave = tid >> 5;

  __shared__ float Etr[32][33];          // exp(trans*IMP), zero padded
  __shared__ float Mring[32][28];        // ring of cached M-rows
  __shared__ float segbuf[2][32][28];    // early-step seg tiles (q=0..31)
  __shared__ float segtd[2][QQ * LL];    // TDM seg tiles, row j = q=30-j
  __shared__ float earow[32];            // exp(alpha - max), zero padded
  __shared__ float alphaL[28];           // current alpha row
  __shared__ float pmax[4][28], psum[4][28];
  __shared__ float td[LL];
  __shared__ float mnewS;

  const float* __restrict__ segB = seg + (size_t)b * TT * TT * LL;

  // ---------------- init ----------------
  for (int i = tid; i < 32 * 32; i += 128) {
    int lp = i >> 5, l = i & 31;
    Etr[lp][l] = (lp < LL && l < LL) ? __expf(trans[lp * LL + l] * IMPF) : 0.f;
  }
  if (tid < LL) {
    td[tid] = trans[tid * LL + tid];
    alphaL[tid] = segB[tid];                       // alpha0 = seg[b,0,0,:]
  }
  for (int i = tid; i < 32 * LL; i += 128) {       // seg tile for t=1
    int q = i / LL, l = i - q * LL;
    int st = 1 - q; if (st < 0) st = 0;
    segbuf[1][q][l] =
        __builtin_nontemporal_load(&segB[((size_t)st * TT + 1) * LL + l]);
  }
  __syncthreads();

  float pre[7];
  // ---------------- sequential scan ----------------
  for (int t = 1; t < TT; ++t) {
    const int head = (1 - t) & 31;                 // ring row of logical q=0
    const int cur  = t & 1;
    const int msp  = (t + 1 < KK) ? (t + 1) : KK;  // max_span
    const int qmax = (t < QQ) ? t : QQ;
    const bool bigt = (t >= 33);                   // this step's tile via TDM

    // commit register-prefetched early tile for this step (t = 2..32)
    if (t >= 2 && t <= 32) {
#pragma unroll
      for (int j = 0; j < 7; ++j) {
        int i = tid + j * 128;
        if (i < 32 * LL) segbuf[cur][i / LL][i - (i / LL) * LL] = pre[j];
      }
    }
    // wait for this step's TDM tile, then kick off next step's TDM tile
    if (bigt) __builtin_amdgcn_s_wait_tensorcnt(0);
    if (wave == 0 && t >= 32 && t + 1 < TT) {
      const int u = t + 1;                          // rows start = u-30+j
      tdm_load_tile31x26(&segB[((size_t)(u - 30) * TT + u) * LL],
                         (unsigned)(uintptr_t)&segtd[u & 1][0]);
    }
    // register-prefetch next early tile (u = t+1 <= 32), clamped starts
    if (t <= 31) {
      const int u = t + 1;
#pragma unroll
      for (int j = 0; j < 7; ++j) {
        int i = tid + j * 128;
        float v = 0.f;
        if (i < 32 * LL) {
          int q = i / LL, l = i - q * LL;
          int st = u - q; if (st < 0) st = 0;
          v = __builtin_nontemporal_load(&segB[((size_t)st * TT + u) * LL + l]);
        }
        pre[j] = v;
      }
    }

    // exp row of alpha_{t-1}
    if (tid < 32) {
      float mn = NEGF;
      for (int l = 0; l < LL; ++l) mn = fmaxf(mn, alphaL[l]);
      earow[tid] = (tid < LL) ? __expf(alphaL[tid] - mn) : 0.f;
      if (tid == 0) mnewS = mn;
    }
    __syncthreads();

    // M-row of alpha_{t-1} via WMMA: (1x32) @ Etr(32x32); wave nt covers
    // output cols 16*nt..16*nt+15; only A-row M=0 is nonzero.
    // Unconditional LDS broadcasts + v_cndmask (no exec juggling); two
    // interleaved accumulators to halve the WMMA RAW chain.
    if (wave < 2) {
      const int nt = wave;
      const bool row0 = (lane & 15) == 0;
      v8f acc0 = {}, acc1 = {};
#pragma unroll
      for (int kc = 0; kc < 8; ++kc) {
        const int k0 = kc * 4 + ((lane >> 4) << 1);
        const float e0 = earow[k0];
        const float e1 = earow[k0 + 1];
        v2f a, bf;
        a[0] = row0 ? e0 : 0.f;
        a[1] = row0 ? e1 : 0.f;
        const int nc = (lane & 15) + 16 * nt;
        bf[0] = Etr[k0][nc];
        bf[1] = Etr[k0 + 1][nc];
        if (kc & 1)
          acc1 = __builtin_amdgcn_wmma_f32_16x16x4_f32(
              false, a, false, bf, (short)0, acc1, false, false);
        else
          acc0 = __builtin_amdgcn_wmma_f32_16x16x4_f32(
              false, a, false, bf, (short)0, acc0, false, false);
      }
      if (lane < 16) {
        int n = 16 * nt + lane;
        if (n < LL) Mring[head][n] = mnewS + __logf(acc0[0] + acc1[0]);
      }
    }
    __syncthreads();

    // outer LSE over q: 26 labels x 4 q-groups of 8; pass 1 = partial max
    if (tid < 104) {
      const int g = tid / LL, l = tid - g * LL;
      const int q0 = g * 8;
      const int q1 = (q0 + 8 < qmax) ? q0 + 8 : qmax;
      const float tdl = td[l];
      float pm = NEGF;
      for (int q = q0; q < q1; ++q) {
        float sv = bigt ? segtd[cur][(30 - q) * LL + l] : segbuf[cur][q][l];
        float val = Mring[(head + q) & 31][l] + sv * (float)(msp - 1 - q)
                  + tdl * (float)(msp - 2 - q);
        pm = fmaxf(pm, val);
      }
      if (g == 0 && (t + 1 <= KK)) {    // score0: q==t row holds seg[b,0,t,:]
        float sc0 = segbuf[cur][t][l] * (float)(t + 1) + tdl * ((float)t * IMPF);
        pm = fmaxf(pm, sc0);
      }
      pmax[g][l] = pm;
    }
    __syncthreads();

    // pass 2 = partial sums against the global max
    if (tid < 104) {
      const int g = tid / LL, l = tid - g * LL;
      const int q0 = g * 8;
      const int q1 = (q0 + 8 < qmax) ? q0 + 8 : qmax;
      const float tdl = td[l];
      const float gm = fmaxf(fmaxf(pmax[0][l], pmax[1][l]),
                             fmaxf(pmax[2][l], pmax[3][l]));
      float ps = 0.f;
      for (int q = q0; q < q1; ++q) {
        float sv = bigt ? segtd[cur][(30 - q) * LL + l] : segbuf[cur][q][l];
        float val = Mring[(head + q) & 31][l] + sv * (float)(msp - 1 - q)
                  + tdl * (float)(msp - 2 - q);
        ps += __expf(val - gm);
      }
      if (g == 0 && (t + 1 <= KK)) {
        float sc0 = segbuf[cur][t][l] * (float)(t + 1) + tdl * ((float)t * IMPF);
        ps += __expf(sc0 - gm);
      }
      psum[g][l] = ps;
    }
    __syncthreads();

    // combine -> alpha_t
    if (tid < LL) {
      const float gm = fmaxf(fmaxf(pmax[0][tid], pmax[1][tid]),
                             fmaxf(pmax[2][tid], pmax[3][tid]));
      const float s = psum[0][tid] + psum[1][tid] + psum[2][tid] + psum[3][tid];
      alphaL[tid] = gm + __logf(s);
    }
    __syncthreads();
  }

  // log_z = LSE_l alpha_{T-1}[l]
  if (tid == 0) {
    float mx = NEGF;
    for (int l = 0; l < LL; ++l) mx = fmaxf(mx, alphaL[l]);
    float s = 0.f;
    for (int l = 0; l < LL; ++l) s += __expf(alphaL[l] - mx);
    out[b] = mx + __logf(s);
  }
}

extern "C" void kernel_launch(void* const* d_in, const int* in_sizes, int n_in,
                              void* d_out, int out_size, void* d_ws, size_t ws_size,
                              hipStream_t stream) {
  const float* seg   = (const float*)d_in[0];   // (B,500,500,26) f32
  const float* trans = (const float*)d_in[1];   // (26,26) f32
  float* out = (float*)d_out;                   // (B,) f32
  const int B = in_sizes[0] / (TT * TT * LL);
  semicrf_fwd_kernel<<<dim3(B), dim3(128), 0, stream>>>(seg, trans, out);
}